// TransportModule_7344394076294
// MI455X (gfx1250) — compile-verified
//
#include <hip/hip_runtime.h>
#include <hip/hip_bf16.h>
#include <math.h>

typedef __attribute__((ext_vector_type(2))) float v2f;
typedef __attribute__((ext_vector_type(8))) float v8f;

#define NPTS 16384   // N
#define BB   4       // B
#define DD   128     // D
#define PP   100     // P
#define PPAD 112     // P padded to 7 tiles of 16

// ---------------------------------------------------------------------------
// Kernel 1: normalize theta rows (L2, eps 1e-12). Rows [100,112) zeroed so the
// padded p-tile of the projection WMMA reads clean zeros.
// ---------------------------------------------------------------------------
__global__ __launch_bounds__(DD) void k_norm(const float* __restrict__ th,
                                             float* __restrict__ thN) {
    const int p = blockIdx.x;
    const int d = threadIdx.x;
    __shared__ float red[DD];
    float v = (p < PP) ? th[p * DD + d] : 0.0f;
    red[d] = v * v;
    __syncthreads();
    for (int s = DD / 2; s > 0; s >>= 1) {
        if (d < s) red[d] += red[d + s];
        __syncthreads();
    }
    float nrm = fmaxf(sqrtf(red[0]), 1e-12f);
    thN[p * DD + d] = (p < PP) ? (v / nrm) : 0.0f;
}

// ---------------------------------------------------------------------------
// Kernel 2: projections  proj[p,b,n] = sum_d src[b,n,d] * thetaN[p,d]
// One WG per (n-block of 128, b, src∈{x,y}). X-block (128x128 f32) and the
// whole theta matrix staged in LDS (padded stride 132 => conflict-free banks).
// WMMA tile: M=p (16), N=n (16 lanes -> coalesced stores), K=d (32 x K=4).
// ---------------------------------------------------------------------------
__global__ __launch_bounds__(256) void k_proj(const float* __restrict__ x,
                                              const float* __restrict__ y,
                                              const float* __restrict__ thN,
                                              float* __restrict__ xp,
                                              float* __restrict__ yp) {
    const int nbase = blockIdx.x * 128;
    const int b     = blockIdx.y;
    const float* src = blockIdx.z ? y : x;
    float*       dst = blockIdx.z ? yp : xp;

    __shared__ float sT[PPAD * 132];   // theta, row stride 132 dwords
    __shared__ float sX[128 * 132];    // 128 points x 128 dims

    const int tid = threadIdx.x;
    for (int i = tid; i < PPAD * DD; i += 256)
        sT[(i >> 7) * 132 + (i & 127)] = thN[i];
    const float* srcb = src + ((size_t)b * NPTS + nbase) * DD;
    for (int i = tid; i < 128 * DD; i += 256)
        sX[(i >> 7) * 132 + (i & 127)] = srcb[i];
    __syncthreads();

    const int wave = tid >> 5;
    const int lane = tid & 31;
    const int ln   = lane & 15;
    const bool hi  = lane >= 16;
    const int nrow = wave * 16 + ln;          // row inside sX
    const int ngl  = nbase + nrow;            // global n

    for (int pt = 0; pt < 7; ++pt) {
        const int p0 = pt * 16;
        v8f c = {};
#pragma unroll
        for (int kk = 0; kk < 32; ++kk) {
            const int dk = kk * 4 + (hi ? 2 : 0);
            v2f a  = *(const v2f*)&sT[(p0 + ln) * 132 + dk];  // A[p][K], A[p][K+1]
            v2f bb = *(const v2f*)&sX[nrow * 132 + dk];       // B[K][n], B[K+1][n]
            c = __builtin_amdgcn_wmma_f32_16x16x4_f32(
                    false, a, false, bb, (short)0, c, false, false);
        }
        const int prow = p0 + (hi ? 8 : 0);
#pragma unroll
        for (int v = 0; v < 8; ++v) {
            const int p = prow + v;
            if (p < PP)
                dst[(size_t)p * (BB * NPTS) + (size_t)b * NPTS + ngl] = c[v];
        }
    }
}

// ---------------------------------------------------------------------------
// Kernel 3: per-(p,b) bitonic sort of 16384 keys fully in LDS (192 KB).
// y: keys only. x: keys + original index (gives argsort). The two sort
// networks share one barrier per pass. Then diff is scattered in LDS so the
// global write of dp[p,b,n] = y_sorted[rank_x(n)] - x_proj[n] is coalesced.
// ---------------------------------------------------------------------------
__global__ __launch_bounds__(1024) void k_sort(const float* __restrict__ xp,
                                               const float* __restrict__ yp,
                                               float* __restrict__ dp) {
    const int p = blockIdx.x >> 2;
    const int b = blockIdx.x & 3;
    const size_t off = ((size_t)p * BB + b) * NPTS;

    __shared__ float ykey[NPTS];   // 64 KB
    __shared__ float xkey[NPTS];   // 64 KB
    __shared__ int   xidx[NPTS];   // 64 KB

    const int tid = threadIdx.x;
    for (int i = tid; i < NPTS; i += 1024) {
        ykey[i] = yp[off + i];
        xkey[i] = xp[off + i];
        xidx[i] = i;
    }
    __syncthreads();

    for (int k = 2; k <= NPTS; k <<= 1) {
        for (int j = k >> 1; j > 0; j >>= 1) {
            for (int i = tid; i < NPTS; i += 1024) {
                const int ixj = i ^ j;
                if (ixj > i) {
                    const bool up = ((i & k) == 0);
                    float ya = ykey[i], yb = ykey[ixj];
                    if ((ya > yb) == up) { ykey[i] = yb; ykey[ixj] = ya; }
                    float xa = xkey[i], xb = xkey[ixj];
                    if ((xa > xb) == up) {
                        xkey[i] = xb; xkey[ixj] = xa;
                        int t = xidx[i]; xidx[i] = xidx[ixj]; xidx[ixj] = t;
                    }
                }
            }
            __syncthreads();
        }
    }

    // diff at sorted position j belongs to original index xidx[j]
    float dv[NPTS / 1024];
    int   di[NPTS / 1024];
    int c = 0;
    for (int i = tid; i < NPTS; i += 1024) {
        dv[c] = ykey[i] - xkey[i];
        di[c] = xidx[i];
        ++c;
    }
    __syncthreads();
    c = 0;
    for (int i = tid; i < NPTS; i += 1024) { ykey[di[c]] = dv[c]; ++c; }
    __syncthreads();
    for (int i = tid; i < NPTS; i += 1024) dp[off + i] = ykey[i];
}

// ---------------------------------------------------------------------------
// Kernel 4: out[b,n,d] = x[b,n,d] + (1/P) * sum_p dp[p,b,n] * thetaN[p,d]
// WMMA tile: M=n (16), N=d (16 lanes -> coalesced), K=p (exactly 25 x K=4).
// A (dp^T) hoisted into 50 VGPRs per wave and reused across all 8 d-tiles.
// Theta staged in LDS with stride 136 (lo/hi half-waves on disjoint banks).
// ---------------------------------------------------------------------------
__global__ __launch_bounds__(256) void k_out(const float* __restrict__ x,
                                             const float* __restrict__ thN,
                                             const float* __restrict__ dp,
                                             float* __restrict__ out) {
    const int nbase = blockIdx.x * 128;
    const int b     = blockIdx.y;

    __shared__ float sT[PP * 136];
    const int tid = threadIdx.x;
    for (int i = tid; i < PP * DD; i += 256)
        sT[(i >> 7) * 136 + (i & 127)] = thN[i];
    __syncthreads();

    const int wave = tid >> 5;
    const int lane = tid & 31;
    const int ln   = lane & 15;
    const bool hi  = lane >= 16;
    const int nr   = nbase + wave * 16 + ln;       // n column for A loads
    const size_t BN   = (size_t)BB * NPTS;
    const size_t boff = (size_t)b * NPTS + nr;

    v2f a[25];
#pragma unroll
    for (int kk = 0; kk < 25; ++kk) {
        const int pk = kk * 4 + (hi ? 2 : 0);
        a[kk].x = dp[(size_t)pk * BN + boff];
        a[kk].y = dp[(size_t)(pk + 1) * BN + boff];
    }

    for (int dt = 0; dt < 8; ++dt) {
        const int d0 = dt * 16;
        v8f c = {};
#pragma unroll
        for (int kk = 0; kk < 25; ++kk) {
            const int pk = kk * 4 + (hi ? 2 : 0);
            v2f bb;
            bb.x = sT[pk * 136 + d0 + ln];
            bb.y = sT[(pk + 1) * 136 + d0 + ln];
            c = __builtin_amdgcn_wmma_f32_16x16x4_f32(
                    false, a[kk], false, bb, (short)0, c, false, false);
        }
#pragma unroll
        for (int v = 0; v < 8; ++v) {
            const int nrow = nbase + wave * 16 + v + (hi ? 8 : 0);
            const size_t idx = ((size_t)b * NPTS + nrow) * DD + d0 + ln;
            out[idx] = x[idx] + 0.01f * c[v];   // 1/P = 0.01
        }
    }
}

// ---------------------------------------------------------------------------
extern "C" void kernel_launch(void* const* d_in, const int* in_sizes, int n_in,
                              void* d_out, int out_size, void* d_ws, size_t ws_size,
                              hipStream_t stream) {
    const float* x  = (const float*)d_in[0];
    const float* y  = (const float*)d_in[1];
    const float* th = (const float*)d_in[2];
    float* out = (float*)d_out;

    float* ws   = (float*)d_ws;
    float* thN  = ws;                                   // PPAD*DD
    float* xp   = thN + (size_t)PPAD * DD;              // PP*B*N
    float* yp   = xp + (size_t)PP * BB * NPTS;          // PP*B*N
    float* dp   = yp + (size_t)PP * BB * NPTS;          // PP*B*N

    k_norm<<<dim3(PPAD), dim3(DD), 0, stream>>>(th, thN);
    k_proj<<<dim3(NPTS / 128, BB, 2), dim3(256), 0, stream>>>(x, y, thN, xp, yp);
    k_sort<<<dim3(PP * BB), dim3(1024), 0, stream>>>(xp, yp, dp);
    k_out<<<dim3(NPTS / 128, BB), dim3(256), 0, stream>>>(x, thN, dp, out);
}